// SimpleLSTM_46076409151800
// MI455X (gfx1250) — compile-verified
//
#include <hip/hip_runtime.h>

// ---------------------------------------------------------------------------
// LSTM (T=2048, B=1024, I=4, H=10, O=4) on gfx1250 using V_WMMA_F32_16X16X4_F32.
//
// One wave (32 lanes) per 16-batch block; 64 blocks. Per step:
//   gates[16x64pad] = [h(10) | x(4) | 1] @ Wcat^T, 4 N-tiles (i,f,g,o).
//   Each tile = (2-deep WMMA chain over h chunks, the critical path) +
//   (2-deep chain over x/bias chunks, off-path) summed with v_pk_add_f32.
//   Cell update is per-lane VALU using v_tanh_f32.
//   h round-trips through a bank-padded LDS tile to rebuild the A layout.
//   FC output = 4 more WMMAs reusing the same A regs, pipelined 1 step behind.
//   x loads software-pipelined one iteration ahead + L2 prefetch 8 ahead.
//   a2/a3 construction is branchless (cndmask, no EXEC save/restore).
// ---------------------------------------------------------------------------

typedef float v2f __attribute__((ext_vector_type(2)));
typedef float v8f __attribute__((ext_vector_type(8)));

#define T_STEPS 2048
#define BATCH   1024
#define IN_D    4
#define HID     10
#define OUTD    4

static __device__ __forceinline__ v8f wmma4(v2f a, v2f b, v8f c) {
  // D = A(16x4 f32) * B(4x16 f32) + C(16x16 f32)
  return __builtin_amdgcn_wmma_f32_16x16x4_f32(
      /*neg_a=*/false, a, /*neg_b=*/false, b,
      /*c_mod=*/(short)0, c, /*reuse_a=*/false, /*reuse_b=*/false);
}

// ---- fast activations: keep the recurrence critical path short -----------
static __device__ __forceinline__ float fast_tanh(float v) {
#if __has_builtin(__builtin_amdgcn_tanhf)
  return __builtin_amdgcn_tanhf(v);               // v_tanh_f32
#elif __has_builtin(__builtin_amdgcn_tanh_f32)
  return __builtin_amdgcn_tanh_f32(v);
#else
  const float e = __expf(-2.0f * v);              // v_exp_f32
  return (1.0f - e) * __builtin_amdgcn_rcpf(1.0f + e);
#endif
}
static __device__ __forceinline__ float fast_sig(float v) {
#if __has_builtin(__builtin_amdgcn_tanhf) || __has_builtin(__builtin_amdgcn_tanh_f32)
  return __builtin_fmaf(0.5f, fast_tanh(0.5f * v), 0.5f);
#else
  return __builtin_amdgcn_rcpf(1.0f + __expf(-v));
#endif
}

__global__ __launch_bounds__(32) void lstm_wmma_kernel(
    const float* __restrict__ x,    const float* __restrict__ h0,
    const float* __restrict__ c0,   const float* __restrict__ W_ih,
    const float* __restrict__ W_hh, const float* __restrict__ b_ih,
    const float* __restrict__ b_hh, const float* __restrict__ W_fc,
    const float* __restrict__ b_fc, float* __restrict__ out)
{
  const int lane = threadIdx.x;     // 0..31
  const int n    = lane & 15;       // tile column (N) / A-row (M)
  const int half = lane >> 4;       // 0: K rows {0,1}+4c, 1: K rows {2,3}+4c
  const int mb   = blockIdx.x << 4; // batch base for this wave

  // ---- Resident B-matrix weight tiles ------------------------------------
  // Combined K axis: k<10 -> W_hh[g][k]; 10<=k<14 -> W_ih[g][k-10];
  //                  k==14 -> b_ih[g]+b_hh[g] (bias row, A supplies 1.0); k==15 -> 0
  v2f Bw[4][4];
  #pragma unroll
  for (int tg = 0; tg < 4; ++tg) {
    #pragma unroll
    for (int c = 0; c < 4; ++c) {
      float vx = 0.0f, vy = 0.0f;
      if (n < HID) {
        const int g  = tg * HID + n;
        const int kx = 4 * c + 2 * half;
        const int ky = kx + 1;
        vx = (kx < 10) ? W_hh[g * 10 + kx]
           : (kx < 14) ? W_ih[g * 4 + (kx - 10)]
           : (kx == 14) ? (b_ih[g] + b_hh[g]) : 0.0f;
        vy = (ky < 10) ? W_hh[g * 10 + ky]
           : (ky < 14) ? W_ih[g * 4 + (ky - 10)]
           : (ky == 14) ? (b_ih[g] + b_hh[g]) : 0.0f;
      }
      Bw[tg][c].x = vx;
      Bw[tg][c].y = vy;
    }
  }
  // FC tiles: N = output channel (n<4); k<10 -> W_fc[n][k]; k==14 -> b_fc[n].
  v2f Bfc[4];
  #pragma unroll
  for (int c = 0; c < 4; ++c) {
    float vx = 0.0f, vy = 0.0f;
    if (n < OUTD) {
      const int kx = 4 * c + 2 * half;
      const int ky = kx + 1;
      vx = (kx < 10) ? W_fc[n * 10 + kx] : (kx == 14) ? b_fc[n] : 0.0f;
      vy = (ky < 10) ? W_fc[n * 10 + ky] : (ky == 14) ? b_fc[n] : 0.0f;
    }
    Bfc[c].x = vx;
    Bfc[c].y = vy;
  }

  // ---- State -------------------------------------------------------------
  __shared__ float hsh[16][17];  // h tile (row m, col u), padded vs bank conflicts
  float cst[8];                  // c state, D-layout: reg r -> (m = r + 8*half, u = n)
  float hreg[8];                 // last h, D-layout (for final hT store)
  #pragma unroll
  for (int r = 0; r < 8; ++r) {
    const int m = r + 8 * half;
    hsh[m][n] = (n < HID) ? h0[(mb + m) * HID + n] : 0.0f;
    cst[r]    = (n < HID) ? c0[(mb + m) * HID + n] : 0.0f;
    hreg[r]   = 0.0f;
  }
  __syncthreads();

  const size_t tailBase = (size_t)T_STEPS * BATCH * OUTD;  // hT/cT region
  const bool lowhalf = (half == 0);

  // x load pipelined one iteration ahead (hide L2 latency; prefetch covers HBM)
  const float* xlane = x + (size_t)(mb + n) * IN_D;
  float4 x4 = *reinterpret_cast<const float4*>(xlane);

  #pragma unroll 1
  for (int t = 0; t < T_STEPS; ++t) {
    const float4 x4cur = x4;
    {
      const int tn = (t + 1 < T_STEPS) ? (t + 1) : (T_STEPS - 1);
      x4 = *reinterpret_cast<const float4*>(xlane + (size_t)tn * BATCH * IN_D);
      const int tp = (t + 8 < T_STEPS) ? (t + 8) : (T_STEPS - 1);  // speculative
      __builtin_prefetch(xlane + (size_t)tp * BATCH * IN_D, 0, 1);
    }

    // A-matrix chunks: lanes0-15 hold K=4c+{0,1}, lanes16-31 K=4c+{2,3}.
    // Branchless: all lanes read hsh[n][8..9]; selects become v_cndmask.
    v2f a0, a1, a2, a3;
    const int kb = 2 * half;
    a0.x = hsh[n][0 + kb]; a0.y = hsh[n][1 + kb];
    a1.x = hsh[n][4 + kb]; a1.y = hsh[n][5 + kb];
    const float h8 = hsh[n][8];
    const float h9 = hsh[n][9];
    a2.x = lowhalf ? h8 : x4cur.x;
    a2.y = lowhalf ? h9 : x4cur.y;
    a3.x = lowhalf ? x4cur.z : 1.0f;
    a3.y = lowhalf ? x4cur.w : 0.0f;

    // ---- Gate GEMM: per tile, h-chain (critical) + x/bias-chain (off-path)
    v8f z = {0.0f, 0.0f, 0.0f, 0.0f, 0.0f, 0.0f, 0.0f, 0.0f};
    v8f Dx0 = wmma4(a3, Bw[0][3], wmma4(a2, Bw[0][2], z));
    v8f Dx1 = wmma4(a3, Bw[1][3], wmma4(a2, Bw[1][2], z));
    v8f Dx2 = wmma4(a3, Bw[2][3], wmma4(a2, Bw[2][2], z));
    v8f Dx3 = wmma4(a3, Bw[3][3], wmma4(a2, Bw[3][2], z));
    v8f Dh0 = wmma4(a1, Bw[0][1], wmma4(a0, Bw[0][0], z));
    v8f Dh1 = wmma4(a1, Bw[1][1], wmma4(a0, Bw[1][0], z));
    v8f Dh2 = wmma4(a1, Bw[2][1], wmma4(a0, Bw[2][0], z));
    v8f Dh3 = wmma4(a1, Bw[3][1], wmma4(a0, Bw[3][0], z));

    // ---- FC output for h_{t-1} (pipelined; a-regs hold h_{t-1}) ----------
    if (t > 0) {
      v8f of = wmma4(a0, Bfc[0], z);
      of = wmma4(a1, Bfc[1], of);
      of = wmma4(a2, Bfc[2], of);
      of = wmma4(a3, Bfc[3], of);
      if (n < OUTD) {
        #pragma unroll
        for (int r = 0; r < 8; ++r) {
          const int m = r + 8 * half;
          out[((size_t)(t - 1) * BATCH + mb + m) * OUTD + n] = of[r];
        }
      }
    }

    const v8f D0 = Dh0 + Dx0;   // v_pk_add_f32 x4 per tile
    const v8f D1 = Dh1 + Dx1;
    const v8f D2 = Dh2 + Dx2;
    const v8f D3 = Dh3 + Dx3;

    // ---- Cell update (pure per-lane) + write h tile back to LDS ----------
    #pragma unroll
    for (int r = 0; r < 8; ++r) {
      const float ig = fast_sig(D0[r]);
      const float fg = fast_sig(D1[r]);
      const float gg = fast_tanh(D2[r]);
      const float og = fast_sig(D3[r]);
      const float cc = fg * cst[r] + ig * gg;
      cst[r] = cc;
      const float hh = og * fast_tanh(cc);
      hreg[r] = hh;
      hsh[r + 8 * half][n] = hh;
    }
    __syncthreads();
  }

  // ---- FC for the final h_{T-1} -------------------------------------------
  {
    v2f a0, a1, a2, a3;
    const int kb = 2 * half;
    a0.x = hsh[n][0 + kb]; a0.y = hsh[n][1 + kb];
    a1.x = hsh[n][4 + kb]; a1.y = hsh[n][5 + kb];
    const float h8 = hsh[n][8];
    const float h9 = hsh[n][9];
    a2.x = lowhalf ? h8 : 0.0f;
    a2.y = lowhalf ? h9 : 0.0f;
    a3.x = lowhalf ? 0.0f : 1.0f;
    a3.y = 0.0f;
    v8f z = {0.0f, 0.0f, 0.0f, 0.0f, 0.0f, 0.0f, 0.0f, 0.0f};
    v8f of = wmma4(a3, Bfc[3], wmma4(a2, Bfc[2], wmma4(a1, Bfc[1], wmma4(a0, Bfc[0], z))));
    if (n < OUTD) {
      #pragma unroll
      for (int r = 0; r < 8; ++r) {
        const int m = r + 8 * half;
        out[((size_t)(T_STEPS - 1) * BATCH + mb + m) * OUTD + n] = of[r];
      }
    }
  }

  // ---- hT / cT tail outputs ----------------------------------------------
  if (n < HID) {
    #pragma unroll
    for (int r = 0; r < 8; ++r) {
      const int b = mb + r + 8 * half;
      out[tailBase + (size_t)b * HID + n]                       = hreg[r];
      out[tailBase + (size_t)BATCH * HID + (size_t)b * HID + n] = cst[r];
    }
  }
}

extern "C" void kernel_launch(void* const* d_in, const int* in_sizes, int n_in,
                              void* d_out, int out_size, void* d_ws, size_t ws_size,
                              hipStream_t stream) {
  (void)in_sizes; (void)n_in; (void)out_size; (void)d_ws; (void)ws_size;
  const float* x    = (const float*)d_in[0];
  const float* h0   = (const float*)d_in[1];
  const float* c0   = (const float*)d_in[2];
  const float* W_ih = (const float*)d_in[3];
  const float* W_hh = (const float*)d_in[4];
  const float* b_ih = (const float*)d_in[5];
  const float* b_hh = (const float*)d_in[6];
  const float* W_fc = (const float*)d_in[7];
  const float* b_fc = (const float*)d_in[8];
  float* out = (float*)d_out;

  lstm_wmma_kernel<<<dim3(BATCH / 16), dim3(32), 0, stream>>>(
      x, h0, c0, W_ih, W_hh, b_ih, b_hh, W_fc, b_fc, out);
}